// BCMSELoss_25907242729551
// MI455X (gfx1250) — compile-verified
//
#include <hip/hip_runtime.h>

typedef __attribute__((ext_vector_type(2))) float v2f;
typedef __attribute__((ext_vector_type(4))) float v4f;
typedef __attribute__((ext_vector_type(8))) float v8f;

// ---------------------------------------------------------------------------
// Wave-wide reduction of two independent f32 accumulators using
// V_WMMA_F32_16X16X4_F32.
//
// A-matrix (16x4 f32) VGPR layout (ISA 7.12.2): lane L<16 holds A[L,0] in
// VGPR0 and A[L,1] in VGPR1; lane 16+L holds A[L,2]/A[L,3]. So per-lane
// a = {sq, pen} gives A[m, 0..3] = {sq[m], pen[m], sq[m+16], pen[m+16]}.
// A uniform B with {VGPR0,VGPR1} = {1,0} selects K in {0,2} -> row sums of sq;
// {0,1} selects K in {1,3} -> row sums of pen, replicated across all N.
// D VGPR j holds row j (lanes 0-15) / row 8+j (lanes 16-31); summing the 8
// D VGPRs leaves half-sum per lane-half, one shfl_xor(16) completes it.
// EXEC is all-ones at every call site (full blocks, converged control flow).
// ---------------------------------------------------------------------------
__device__ __forceinline__ void wave_reduce_pair(float& sq, float& pen) {
#if __has_builtin(__builtin_amdgcn_wmma_f32_16x16x4_f32)
  v2f a;  a.x = sq;    a.y = pen;
  v2f b0; b0.x = 1.0f; b0.y = 0.0f;   // select sq slots (K = 0,2)
  v2f b1; b1.x = 0.0f; b1.y = 1.0f;   // select pen slots (K = 1,3)
  v8f c = {};
  v8f dq = __builtin_amdgcn_wmma_f32_16x16x4_f32(
      false, a, false, b0, (short)0, c, false, false);
  v8f dp = __builtin_amdgcn_wmma_f32_16x16x4_f32(
      false, a, false, b1, (short)0, c, false, false);
  float tq = ((dq[0] + dq[1]) + (dq[2] + dq[3])) +
             ((dq[4] + dq[5]) + (dq[6] + dq[7]));
  float tp = ((dp[0] + dp[1]) + (dp[2] + dp[3])) +
             ((dp[4] + dp[5]) + (dp[6] + dp[7]));
  sq  = tq + __shfl_xor(tq, 16, 32);
  pen = tp + __shfl_xor(tp, 16, 32);
#else
  for (int off = 16; off > 0; off >>= 1) {
    sq  += __shfl_xor(sq, off, 32);
    pen += __shfl_xor(pen, off, 32);
  }
#endif
}

// Plain column (col 0): squared error.
__device__ __forceinline__ void acc_lin(float o, float t, float& sq) {
  float d = o - t;
  sq = fmaf(d, d, sq);
}

// Angular column (cols 1,2): wrap, penalty, circular shift, squared error.
// mod(o,1) == o - floor(o) for finite o (sign of divisor). shift condition:
// d = wrapped - t; |d| > 0.5 with d > 0 <=> t < wrapped -> +1 shift (d -= 1),
// d < -0.5 -> -1 shift (d += 1).
__device__ __forceinline__ void acc_ang(float o, float t, float& sq, float& pen) {
  float f = floorf(o);
  pen += fabsf(f);
  float d = (o - f) - t;
  d = (d > 0.5f) ? (d - 1.0f) : ((d < -0.5f) ? (d + 1.0f) : d);
  sq = fmaf(d, d, sq);
}

// ---------------------------------------------------------------------------
// Pass 1: each thread streams 4-row chunks (12 floats = 3 x b128 per operand,
// 48-byte segments -> 16B aligned, fully coalesced). Column pattern within a
// chunk is compile-time static: i % 3 over i = 0..11. Non-temporal loads keep
// the single-use 201 MB stream from rinsing L2.
// ---------------------------------------------------------------------------
__global__ void __launch_bounds__(256) bcmse_partial(
    const float* __restrict__ outp, const float* __restrict__ tgtp,
    float* __restrict__ partials, long long numChunks, long long nRows) {
  float sq = 0.0f, pen = 0.0f;
  const long long tid    = (long long)blockIdx.x * blockDim.x + threadIdx.x;
  const long long stride = (long long)gridDim.x * blockDim.x;

  for (long long ch = tid; ch < numChunks; ch += stride) {
    const v4f* po = (const v4f*)(outp + ch * 12);
    const v4f* pt = (const v4f*)(tgtp + ch * 12);
    v4f o0 = __builtin_nontemporal_load(po + 0);
    v4f o1 = __builtin_nontemporal_load(po + 1);
    v4f o2 = __builtin_nontemporal_load(po + 2);
    v4f t0 = __builtin_nontemporal_load(pt + 0);
    v4f t1 = __builtin_nontemporal_load(pt + 1);
    v4f t2 = __builtin_nontemporal_load(pt + 2);
    // cols over the 12-float chunk: 0 1 2 | 0 1 2 | 0 1 2 | 0 1 2
    acc_lin(o0[0], t0[0], sq);
    acc_ang(o0[1], t0[1], sq, pen);
    acc_ang(o0[2], t0[2], sq, pen);
    acc_lin(o0[3], t0[3], sq);
    acc_ang(o1[0], t1[0], sq, pen);
    acc_ang(o1[1], t1[1], sq, pen);
    acc_lin(o1[2], t1[2], sq);
    acc_ang(o1[3], t1[3], sq, pen);
    acc_ang(o2[0], t2[0], sq, pen);
    acc_lin(o2[1], t2[1], sq);
    acc_ang(o2[2], t2[2], sq, pen);
    acc_ang(o2[3], t2[3], sq, pen);
  }

  // Tail rows (B not divisible by 4 in the general case): one row per thread.
  const long long tailStart = numChunks * 4;
  const long long nTail     = nRows - tailStart;
  if (tid < nTail) {
    const long long r = tailStart + tid;
    acc_lin(outp[r * 3 + 0], tgtp[r * 3 + 0], sq);
    acc_ang(outp[r * 3 + 1], tgtp[r * 3 + 1], sq, pen);
    acc_ang(outp[r * 3 + 2], tgtp[r * 3 + 2], sq, pen);
  }

  // 32-lane WMMA reduction, then 8-wave LDS combine (256 threads = 8 waves).
  wave_reduce_pair(sq, pen);

  __shared__ float s_sq[8];
  __shared__ float s_pen[8];
  const int wave = threadIdx.x >> 5;
  if ((threadIdx.x & 31) == 0) { s_sq[wave] = sq; s_pen[wave] = pen; }
  __syncthreads();
  if (threadIdx.x == 0) {
    float a = 0.0f, b = 0.0f;
    for (int i = 0; i < 8; ++i) { a += s_sq[i]; b += s_pen[i]; }
    partials[2 * blockIdx.x + 0] = a;   // every block writes -> deterministic
    partials[2 * blockIdx.x + 1] = b;
  }
}

// ---------------------------------------------------------------------------
// Pass 2: one block folds the per-block partials and writes the scalar.
// ---------------------------------------------------------------------------
__global__ void __launch_bounds__(256) bcmse_final(
    const float* __restrict__ partials, int nPart, float* __restrict__ result,
    float inv3B, float invB) {
  float sq = 0.0f, pen = 0.0f;
  for (int i = threadIdx.x; i < nPart; i += 256) {
    sq  += partials[2 * i + 0];
    pen += partials[2 * i + 1];
  }
  wave_reduce_pair(sq, pen);

  __shared__ float s_sq[8];
  __shared__ float s_pen[8];
  const int wave = threadIdx.x >> 5;
  if ((threadIdx.x & 31) == 0) { s_sq[wave] = sq; s_pen[wave] = pen; }
  __syncthreads();
  if (threadIdx.x == 0) {
    float a = 0.0f, b = 0.0f;
    for (int i = 0; i < 8; ++i) { a += s_sq[i]; b += s_pen[i]; }
    result[0] = a * inv3B + b * invB;   // mse + constant_penalty
  }
}

extern "C" void kernel_launch(void* const* d_in, const int* in_sizes, int n_in,
                              void* d_out, int out_size, void* d_ws, size_t ws_size,
                              hipStream_t stream) {
  const float* outputs = (const float*)d_in[0];
  const float* targets = (const float*)d_in[1];
  float* result   = (float*)d_out;
  float* partials = (float*)d_ws;

  const long long total     = (long long)in_sizes[0];  // B * 3
  const long long B         = total / 3;
  const long long numChunks = B / 4;                   // 4 rows per chunk

  const int threads = 256;
  int blocks = 2048;                                   // ~524k threads, 4 chunks each
  const long long maxBlocksWs = (long long)(ws_size / (2 * sizeof(float)));
  if (blocks > maxBlocksWs) blocks = (int)maxBlocksWs;
  if (blocks < 1) blocks = 1;

  bcmse_partial<<<blocks, threads, 0, stream>>>(outputs, targets, partials,
                                                numChunks, B);
  bcmse_final<<<1, threads, 0, stream>>>(partials, blocks, result,
                                         1.0f / (3.0f * (float)B),
                                         1.0f / (float)B);
}